// EventADModel_8770323218527
// MI455X (gfx1250) — compile-verified
//
#include <hip/hip_runtime.h>
#include <math.h>

typedef __attribute__((ext_vector_type(2))) float v2f;
typedef __attribute__((ext_vector_type(8))) float v8f;

#define NWAVES   8
#define NTHREADS 256
#define TT       30
#define MPAD     32
#define NSTEPS   4096

// ---------------------------------------------------------------------------
// Pack a row-major weight W[N][K] into WMMA-B fragment order for
// V_WMMA_F32_16X16X4_F32:  fragment(ntile,ks) is 64 floats;
// value at [frag*64 + lane*2 + j] = W[n][k],  n = ntile*16 + (lane&15),
// k = ks*4 + j + 2*(lane>=16).
// ---------------------------------------------------------------------------
__global__ void pack_b_kernel(const float* __restrict__ W, float* __restrict__ P,
                              int N, int K) {
  int idx = blockIdx.x * blockDim.x + threadIdx.x;
  int total = N * K;
  if (idx >= total) return;
  int frag = idx >> 6;
  int rem  = idx & 63;
  int lane = rem >> 1;
  int j    = rem & 1;
  int kS   = K >> 2;
  int ntile = frag / kS;
  int ks    = frag - ntile * kS;
  int n = ntile * 16 + (lane & 15);
  int k = ks * 4 + j + 2 * (lane >> 4);
  P[idx] = W[(size_t)n * K + k];
}

// ---------------------------------------------------------------------------
// Accumulate one packed weight source into the two M-tile accumulators.
// A is LDS, row-major [32][sA]; B is the packed global buffer.
// ---------------------------------------------------------------------------
__device__ __forceinline__ void gemm_src_acc(
    const float* __restrict__ P, int tileOff, int K,
    const float* __restrict__ A, int sA,
    int nt, int lane, v8f& acc0, v8f& acc1)
{
  const int lo = lane & 15, hi = lane >> 4;
  const int kS = K >> 2;
  const float* Pf = P + ((size_t)(tileOff + nt) * kS) * 64 + lane * 2;
  for (int ks = 0; ks < kS; ++ks) {
    v2f bfrag; bfrag.x = Pf[0]; bfrag.y = Pf[1]; Pf += 64;
    const int kb = ks * 4 + 2 * hi;
    const float* a0p = A + lo * sA + kb;
    v2f a0; a0.x = a0p[0]; a0.y = a0p[1];
    const float* a1p = a0p + 16 * sA;
    v2f a1; a1.x = a1p[0]; a1.y = a1p[1];
    acc0 = __builtin_amdgcn_wmma_f32_16x16x4_f32(false, a0, false, bfrag,
                                                 (short)0, acc0, false, false);
    acc1 = __builtin_amdgcn_wmma_f32_16x16x4_f32(false, a1, false, bfrag,
                                                 (short)0, acc1, false, false);
  }
}

// O[32 rows][cols colOff .. colOff+16*nTiles) = bias + A0*W0^T (+ A1*W1^T)
// Each wave owns N-tiles wid, wid+8, ... Full wave executes (EXEC all ones).
__device__ __forceinline__ void wave_gemm(
    const float* P0, int tileOff0, int K0, const float* A0, int sA0,
    const float* P1, int tileOff1, int K1, const float* A1, int sA1,
    const float* bias,
    float* O, int sO, int colOff, int nTiles, int lane, int wid)
{
  const int lo = lane & 15, hi = lane >> 4;
  for (int nt = wid; nt < nTiles; nt += NWAVES) {
    v8f acc0 = {}; v8f acc1 = {};
    if (bias) {
      float bv = bias[nt * 16 + lo];
#pragma unroll
      for (int r = 0; r < 8; ++r) { acc0[r] = bv; acc1[r] = bv; }
    }
    gemm_src_acc(P0, tileOff0, K0, A0, sA0, nt, lane, acc0, acc1);
    if (P1) gemm_src_acc(P1, tileOff1, K1, A1, sA1, nt, lane, acc0, acc1);
#pragma unroll
    for (int r = 0; r < 8; ++r) {
      int m   = r + 8 * hi;
      int col = colOff + nt * 16 + lo;
      O[(size_t)m * sO + col]        = acc0[r];
      O[(size_t)(m + 16) * sO + col] = acc1[r];
    }
  }
}

// PyTorch GRU gate combine (r,z,n ordering), event layers (H=256).
__device__ __forceinline__ void gru_combine_ev(
    const float* g_rz, const float* gi_n, const float* gh_n,
    const float* __restrict__ bi, const float* __restrict__ bh,
    const float* hprev, int sHP, float* hout, int tid)
{
  for (int e = tid; e < MPAD * 256; e += NTHREADS) {
    int t = e >> 8, h = e & 255;
    float r = g_rz[(size_t)t * 516 + h]       + bi[h]       + bh[h];
    float z = g_rz[(size_t)t * 516 + 256 + h] + bi[256 + h] + bh[256 + h];
    r = 1.f / (1.f + expf(-r));
    z = 1.f / (1.f + expf(-z));
    float n = tanhf(gi_n[(size_t)t * 260 + h] + bi[512 + h] +
                    r * (gh_n[(size_t)t * 260 + h] + bh[512 + h]));
    float hp = hprev[(size_t)t * sHP + h];
    hout[(size_t)t * 260 + h] = (1.f - z) * n + z * hp;
  }
}

struct KParams {
  const float* feat; const float* coord; const int* labels; const unsigned char* mask;
  const float *bi0, *bh0, *bi1, *bh1;
  const float *Wic, *Whc, *bic, *bhc;
  const float *be, *bc, *bf1, *Wf2, *bf2, *wev, *wco;
  const float *PWi0, *PWh0, *PWi1, *PWh1, *PWe, *PWc, *PWf1;
  float* out;
};

// ---------------------------------------------------------------------------
// Persistent single-workgroup kernel: iterates all 4096 serial steps, keeping
// hidden state and all per-step activations in the WGP's 320KB LDS.
// ---------------------------------------------------------------------------
__global__ __launch_bounds__(NTHREADS, 1)
void event_ad_main(KParams p)
{
  extern __shared__ float sm[];
  float* HEV0 = sm;                 // [32][260] persistent h_ev layer0
  float* HEV1 = HEV0 + 32 * 260;    // [32][260] persistent h_ev layer1
  float* HCO  = HEV1 + 32 * 260;    // [32][36]  persistent h_co
  float* XB   = HCO  + 32 * 36;     // [32][68]  feat staging
  float* CO   = XB   + 32 * 68;     // [32][4]   coord staging
  float* GRZ  = CO   + 32 * 4;      // [32][516] r,z pre-gates   (reused: comb)
  float* GIN  = GRZ  + 32 * 516;    // [32][260] x-part n gate   (reused: f1)
  float* GHN  = GIN  + 32 * 260;    // [32][260] h-part n gate
  float* H0   = GHN  + 32 * 260;    // [32][260] fresh layer0 hidden
  float* H1   = H0   + 32 * 260;    // [32][260] fresh layer1 hidden
  float* HCN  = H1   + 32 * 260;    // [32][36]  fresh coord hidden
  float* SATT = HCN  + 32 * 36;     // [3][32]   attention scores (ev0, ev1, co)
  float* AATT = SATT + 96;          // [3][32]   attention weights
  float* LOGI = AATT + 96;          // [32][2]   logits
  float* CE   = LOGI + 64;          // [32]      per-track CE
  __shared__ int labL[32];
  __shared__ int maskL[32];

  const int tid  = threadIdx.x;
  const int lane = tid & 31;
  const int wid  = tid >> 5;
  float lossAcc = 0.f;

  for (int e = tid; e < 32 * 260; e += NTHREADS) { HEV0[e] = 0.f; HEV1[e] = 0.f; }
  for (int e = tid; e < 32 * 36;  e += NTHREADS) HCO[e] = 0.f;
  __syncthreads();

  for (int b = 0; b < NSTEPS; ++b) {
    // ---- load step inputs ----
    for (int e = tid; e < 32 * 64; e += NTHREADS) {
      int t = e >> 6, k = e & 63;
      XB[t * 68 + k] = (t < TT) ? p.feat[((size_t)b * TT + t) * 64 + k] : 0.f;
    }
    for (int e = tid; e < 32 * 4; e += NTHREADS) {
      int t = e >> 2, k = e & 3;
      CO[t * 4 + k] = (t < TT) ? p.coord[((size_t)b * TT + t) * 4 + k] : 0.f;
    }
    if (tid < 32) {
      labL[tid]  = (tid < TT) ? p.labels[(size_t)b * TT + tid] : 0;
      maskL[tid] = (tid < TT) ? (int)p.mask[(size_t)b * TT + tid] : 0;
    }
    __syncthreads();

    // ---- layer 0 GEMMs (fused x*Wi + h*Wh for r,z; split for n) ----
    wave_gemm(p.PWi0, 0, 64, XB, 68,   p.PWh0, 0, 256, HEV0, 260,
              nullptr, GRZ, 516, 0, 32, lane, wid);
    wave_gemm(p.PWi0, 32, 64, XB, 68,  nullptr, 0, 0, nullptr, 0,
              nullptr, GIN, 260, 0, 16, lane, wid);
    wave_gemm(p.PWh0, 32, 256, HEV0, 260, nullptr, 0, 0, nullptr, 0,
              nullptr, GHN, 260, 0, 16, lane, wid);

    // ---- coord GRU (HC=32, K=4/32 -> scalar VALU) ----
    for (int e = tid; e < 32 * 32; e += NTHREADS) {
      int t = e >> 5, j = e & 31;
      float ir = p.bic[j], iz = p.bic[32 + j], inn = p.bic[64 + j];
#pragma unroll
      for (int k = 0; k < 4; ++k) {
        float c = CO[t * 4 + k];
        ir  += c * p.Wic[j * 4 + k];
        iz  += c * p.Wic[(32 + j) * 4 + k];
        inn += c * p.Wic[(64 + j) * 4 + k];
      }
      float hr = p.bhc[j], hz = p.bhc[32 + j], hn = p.bhc[64 + j];
      for (int k = 0; k < 32; ++k) {
        float hv = HCO[t * 36 + k];
        hr += hv * p.Whc[j * 32 + k];
        hz += hv * p.Whc[(32 + j) * 32 + k];
        hn += hv * p.Whc[(64 + j) * 32 + k];
      }
      float r = 1.f / (1.f + expf(-(ir + hr)));
      float z = 1.f / (1.f + expf(-(iz + hz)));
      float n = tanhf(inn + r * hn);
      HCN[t * 36 + j] = (1.f - z) * n + z * HCO[t * 36 + j];
    }
    __syncthreads();

    gru_combine_ev(GRZ, GIN, GHN, p.bi0, p.bh0, HEV0, 260, H0, tid);
    __syncthreads();

    // ---- layer 1 GEMMs ----
    wave_gemm(p.PWi1, 0, 256, H0, 260, p.PWh1, 0, 256, HEV1, 260,
              nullptr, GRZ, 516, 0, 32, lane, wid);
    wave_gemm(p.PWi1, 32, 256, H0, 260, nullptr, 0, 0, nullptr, 0,
              nullptr, GIN, 260, 0, 16, lane, wid);
    wave_gemm(p.PWh1, 32, 256, HEV1, 260, nullptr, 0, 0, nullptr, 0,
              nullptr, GHN, 260, 0, 16, lane, wid);
    __syncthreads();

    gru_combine_ev(GRZ, GIN, GHN, p.bi1, p.bh1, HEV1, 260, H1, tid);
    __syncthreads();

    // ---- fusion: comb = [h1*We^T + be | hc*Wc^T + bc]  (comb == GRZ) ----
    wave_gemm(p.PWe, 0, 256, H1, 260, nullptr, 0, 0, nullptr, 0,
              p.be, GRZ, 516, 0, 16, lane, wid);
    wave_gemm(p.PWc, 0, 32, HCN, 36, nullptr, 0, 0, nullptr, 0,
              p.bc, GRZ, 516, 256, 16, lane, wid);
    __syncthreads();

    // ---- f1 = comb*Wf1^T + bf1 (relu applied on read) -> GIN ----
    wave_gemm(p.PWf1, 0, 512, GRZ, 516, nullptr, 0, 0, nullptr, 0,
              p.bf1, GIN, 260, 0, 16, lane, wid);
    __syncthreads();

    // ---- logits (N=2, scalar) + attention scores (wave reductions) ----
    if (tid < 64) {
      int t = tid >> 1, c = tid & 1;
      float s = p.bf2[c];
      const float* f1row = GIN + t * 260;
      for (int k = 0; k < 256; ++k) s += fmaxf(f1row[k], 0.f) * p.Wf2[c * 256 + k];
      LOGI[t * 2 + c] = s;
    }
    for (int rid = wid; rid < 96; rid += NWAVES) {
      float v;
      if (rid < 64) {
        int l = rid >> 5, t = rid & 31;
        const float* hsrc = (l == 0 ? H0 : H1) + t * 260;
        v = 0.f;
#pragma unroll
        for (int i = 0; i < 8; ++i) { int k = lane + 32 * i; v += tanhf(hsrc[k]) * p.wev[k]; }
      } else {
        int t = rid - 64;
        v = tanhf(HCN[t * 36 + lane]) * p.wco[lane];
      }
      for (int off = 16; off > 0; off >>= 1) v += __shfl_xor(v, off, 32);
      if (lane == 0) {
        if (rid < 64) SATT[(rid >> 5) * 32 + (rid & 31)] = v;
        else          SATT[64 + (rid - 64)] = v;
      }
    }
    __syncthreads();

    // ---- masked softmax (3 rows) ; CE + masked logit output ----
    if (tid < 3) {
      const float* s = SATT + tid * 32;
      float*       a = AATT + tid * 32;
      float mx = -1e30f;
      for (int t = 0; t < TT; ++t) if (maskL[t] && s[t] > mx) mx = s[t];
      float sum = 0.f;
      for (int t = 0; t < TT; ++t) {
        float e = maskL[t] ? expf(s[t] - mx) : 0.f;
        a[t] = e; sum += e;
      }
      float inv = 1.f / fmaxf(sum, 1e-9f);
      for (int t = 0; t < TT; ++t) a[t] *= inv;
    }
    if (tid >= 32 && tid < 32 + TT) {
      int t = tid - 32;
      float l0 = LOGI[t * 2], l1 = LOGI[t * 2 + 1];
      float mx  = fmaxf(l0, l1);
      float lse = mx + logf(expf(l0 - mx) + expf(l1 - mx));
      float ce  = -((labL[t] ? l1 : l0) - lse);
      int m = maskL[t];
      CE[t] = m ? ce : 0.f;
      float* op = p.out + 1 + ((size_t)b * TT + t) * 2;
      op[0] = m ? l0 : 0.f;
      op[1] = m ? l1 : 0.f;
    }
    __syncthreads();

    // ---- loss accumulate + attention-weighted hidden update (mask-gated) ----
    if (tid == 0) for (int t = 0; t < TT; ++t) lossAcc += CE[t];
    for (int e = tid; e < 32 * 256; e += NTHREADS) {
      int t = e >> 8, h = e & 255;
      if (maskL[t]) {
        HEV0[t * 260 + h] = H0[t * 260 + h] * AATT[t];
        HEV1[t * 260 + h] = H1[t * 260 + h] * AATT[32 + t];
      }
    }
    for (int e = tid; e < 32 * 32; e += NTHREADS) {
      int t = e >> 5, j = e & 31;
      if (maskL[t]) HCO[t * 36 + j] = HCN[t * 36 + j] * AATT[64 + t];
    }
    __syncthreads();
  }

  if (tid == 0) p.out[0] = lossAcc;
}

extern "C" void kernel_launch(void* const* d_in, const int* in_sizes, int n_in,
                              void* d_out, int out_size, void* d_ws, size_t ws_size,
                              hipStream_t stream)
{
  (void)in_sizes; (void)n_in; (void)out_size; (void)ws_size;
  const float* feat  = (const float*)d_in[0];
  const float* coord = (const float*)d_in[1];
  const int*   labels = (const int*)d_in[2];
  const unsigned char* mask = (const unsigned char*)d_in[3];
  const float* Wi0 = (const float*)d_in[4];
  const float* Wh0 = (const float*)d_in[5];
  const float* bi0 = (const float*)d_in[6];
  const float* bh0 = (const float*)d_in[7];
  const float* Wi1 = (const float*)d_in[8];
  const float* Wh1 = (const float*)d_in[9];
  const float* bi1 = (const float*)d_in[10];
  const float* bh1 = (const float*)d_in[11];
  const float* Wic = (const float*)d_in[12];
  const float* Whc = (const float*)d_in[13];
  const float* bic = (const float*)d_in[14];
  const float* bhc = (const float*)d_in[15];
  const float* We  = (const float*)d_in[16];
  const float* be  = (const float*)d_in[17];
  const float* Wc  = (const float*)d_in[18];
  const float* bc  = (const float*)d_in[19];
  const float* Wf1 = (const float*)d_in[20];
  const float* bf1 = (const float*)d_in[21];
  const float* Wf2 = (const float*)d_in[22];
  const float* bf2 = (const float*)d_in[23];
  const float* wev = (const float*)d_in[24];
  const float* wco = (const float*)d_in[25];

  // Workspace: packed WMMA-B weights (3.3 MB fp32), L2-resident thereafter.
  float* ws   = (float*)d_ws;
  float* PWi0 = ws;                    // 768*64
  float* PWh0 = PWi0 + 768 * 64;       // 768*256
  float* PWi1 = PWh0 + 768 * 256;
  float* PWh1 = PWi1 + 768 * 256;
  float* PWe  = PWh1 + 768 * 256;      // 256*256
  float* PWc  = PWe  + 256 * 256;      // 256*32
  float* PWf1 = PWc  + 256 * 32;       // 256*512

  auto packLaunch = [&](const float* W, float* P, int N, int K) {
    int total = N * K;
    pack_b_kernel<<<(total + 255) / 256, 256, 0, stream>>>(W, P, N, K);
  };
  packLaunch(Wi0, PWi0, 768, 64);
  packLaunch(Wh0, PWh0, 768, 256);
  packLaunch(Wi1, PWi1, 768, 256);
  packLaunch(Wh1, PWh1, 768, 256);
  packLaunch(We,  PWe,  256, 256);
  packLaunch(Wc,  PWc,  256, 32);
  packLaunch(Wf1, PWf1, 256, 512);

  KParams p { feat, coord, labels, mask,
              bi0, bh0, bi1, bh1,
              Wic, Whc, bic, bhc,
              be, bc, bf1, Wf2, bf2, wev, wco,
              PWi0, PWh0, PWi1, PWh1, PWe, PWc, PWf1,
              (float*)d_out };

  // Dynamic LDS: 63008 floats = 252,032 B (< 320 KB per WGP on CDNA5).
  size_t smemBytes = (size_t)(32*260*2 + 32*36 + 32*68 + 32*4 + 32*516 +
                              32*260*3 + 32*36 + 96 + 96 + 64 + 32) * sizeof(float);
  event_ad_main<<<1, NTHREADS, smemBytes, stream>>>(p);
}